// ImageGradients_76579266887903
// MI455X (gfx1250) — compile-verified
//
#include <hip/hip_runtime.h>
#include <hip/hip_bf16.h>
#include <stdint.h>

// Problem constants (from reference): x is (B=2, C=2, 160,160,160) f32,
// out is (1, B*C=4, 160,160,160) f32 with identical flat layout.
constexpr int Dm = 160, Hm = 160, Wm = 160;
constexpr int BC = 4;
constexpr int HW  = Hm * Wm;
constexpr int DHW = Dm * HW;

// Tile: 8 x 8 x fullW outputs per block; halo of 1 each side.
constexpr int TD = 8, TH = 8;
constexpr int LD = TD + 2;            // 10
constexpr int LH = TH + 2;            // 10
constexpr int LWI = Wm;               // 160 interior columns per LDS row
constexpr int NROWS = LD * LH;        // 100 (d,h) rows in the tile
constexpr int HB = NROWS * LWI;       // halo block base index (16000)
constexpr int NTILES_D = Dm / TD;     // 20
constexpr int NTILES_H = Hm / TH;     // 20
constexpr int WCHUNK = Wm / 4;        // 40 outputs per thread along W

__global__ __launch_bounds__(256)
void sobel3d_mag_kernel(const float* __restrict__ x, float* __restrict__ out) {
    // Interior [10][10][160] + halo [10][10][2] = 16200 floats = 64800 B.
    // Interior rows are 640B-strided -> every 4-float chunk is 16B aligned (b128-legal).
    __shared__ __align__(16) float ldsbuf[HB + NROWS * 2];

    const int tid = threadIdx.x;
    const int blk = blockIdx.x;
    const int bc  = blk / (NTILES_D * NTILES_H);
    const int rem = blk % (NTILES_D * NTILES_H);
    const int h0  = (rem / NTILES_D) * TH;
    const int d0  = (rem % NTILES_D) * TD;

    const unsigned long long chanBase =
        (unsigned long long)(uintptr_t)x + 4ull * ((unsigned long long)bc * DHW);
    const unsigned ldsBase = (unsigned)(uintptr_t)&ldsbuf[0]; // low 32 bits = LDS offset

    // ---------------- Async global -> LDS tile fill (CDNA5 DMA path) ------
    // Interior: 100 rows x 40 b128 chunks (w halo handled separately).
    constexpr int NCH = NROWS * (LWI / 4);   // 4000
    for (int e = tid; e < NCH; e += 256) {
        const int cw = e % (LWI / 4);
        const int r  = e / (LWI / 4);
        const int hy = r % LH;
        const int dz = r / LH;
        int gd = d0 - 1 + dz; gd = gd < 0 ? 0 : (gd > Dm - 1 ? Dm - 1 : gd);
        int gh = h0 - 1 + hy; gh = gh < 0 ? 0 : (gh > Hm - 1 ? Hm - 1 : gh);
        const unsigned long long ga =
            chanBase + 4ull * ((unsigned long long)gd * HW
                               + (unsigned long long)(gh * Wm + cw * 4));
        const unsigned la = ldsBase + 4u * (unsigned)(r * LWI + cw * 4);
        asm volatile("global_load_async_to_lds_b128 %0, %1, off"
                     :: "v"(la), "v"(ga) : "memory");
    }
    // Halo: replicated w=-1 and w=160 columns, 200 b32 ops.
    if (tid < NROWS * 2) {
        const int side = tid & 1;            // 0: w=-1 -> x[...,0], 1: w=160 -> x[...,159]
        const int r    = tid >> 1;
        const int hy = r % LH;
        const int dz = r / LH;
        int gd = d0 - 1 + dz; gd = gd < 0 ? 0 : (gd > Dm - 1 ? Dm - 1 : gd);
        int gh = h0 - 1 + hy; gh = gh < 0 ? 0 : (gh > Hm - 1 ? Hm - 1 : gh);
        const int gw = side ? (Wm - 1) : 0;
        const unsigned long long ga =
            chanBase + 4ull * ((unsigned long long)gd * HW
                               + (unsigned long long)(gh * Wm + gw));
        const unsigned la = ldsBase + 4u * (unsigned)(HB + tid);
        asm volatile("global_load_async_to_lds_b32 %0, %1, off"
                     :: "v"(la), "v"(ga) : "memory");
    }
    asm volatile("s_wait_asynccnt 0" ::: "memory");
    __syncthreads();

    // ---------------- Sliding-window compute ------------------------------
    // 64 (d,h) rows, 4 threads per row each covering 40 consecutive w.
    const int row  = tid >> 2;      // 0..63
    const int subw = tid & 3;       // 0..3
    const int d = row >> 3;         // 0..7   (output d = d0 + d)
    const int h = row & 7;          // 0..7   (output h = h0 + h)
    const int ws = subw * WCHUNK;   // first output w of this thread

    // Per-(dd,dh) row bases. k = dd*3+dh: l0..l2 plane d-1, l3..l5 plane d, l6..l8 plane d+1.
    int baseI[9], baseHalo[9];
    #pragma unroll
    for (int dd = 0; dd < 3; ++dd)
        #pragma unroll
        for (int dh = 0; dh < 3; ++dh) {
            const int r = (d + dd) * LH + (h + dh);
            baseI[dd * 3 + dh]    = r * LWI;
            baseHalo[dd * 3 + dh] = HB + r * 2;
        }

    struct Col { float dpd, dph, c9; };
    auto reduce9 = [&](const float l[9]) -> Col {
        const float pdm = l[0] + l[1] + l[2];       // h-sum, plane d-1
        const float pdp = l[6] + l[7] + l[8];       // h-sum, plane d+1
        const float phm = l[0] + l[3] + l[6];       // d-sum, row h-1
        const float php = l[2] + l[5] + l[8];       // d-sum, row h+1
        Col c;
        c.dpd = pdp - pdm;
        c.dph = php - phm;
        c.c9  = pdm + pdp + (l[3] + l[4] + l[5]);   // full 3x3 (d,h) sum
        return c;
    };
    auto colF = [&](int j) -> Col {                 // j in [0, 159]
        float l[9];
        #pragma unroll
        for (int k = 0; k < 9; ++k) l[k] = ldsbuf[baseI[k] + j];
        return reduce9(l);
    };
    auto colG = [&](int j) -> Col {                 // j in [-1, 160]
        if ((unsigned)j < (unsigned)LWI) return colF(j);
        const int s = (j > 0) ? 1 : 0;
        float l[9];
        #pragma unroll
        for (int k = 0; k < 9; ++k) l[k] = ldsbuf[baseHalo[k] + s];
        return reduce9(l);
    };

    Col a = colG(ws - 1);   // halo only when subw==0
    Col b = colF(ws);

    float* op = out + (size_t)bc * DHW + (size_t)(d0 + d) * HW
                    + (size_t)(h0 + h) * Wm + ws;
    float buf[4];

    auto emit = [&](int i, const Col& c) {
        const float gd = a.dpd + b.dpd + c.dpd;     // Sobel along D
        const float gh = a.dph + b.dph + c.dph;     // Sobel along H
        const float gw = c.c9 - a.c9;               // Sobel along W
        buf[i & 3] = sqrtf(gd * gd + gh * gh + gw * gw);
        if ((i & 3) == 3)
            *reinterpret_cast<float4*>(op + (i - 3)) =
                *reinterpret_cast<const float4*>(buf);   // global_store_b128
        a = b; b = c;
    };

    #pragma unroll 4
    for (int i = 0; i < WCHUNK - 4; ++i)            // i = 0..35: always in range
        emit(i, colF(ws + i + 1));

    #pragma unroll
    for (int i = WCHUNK - 4; i < WCHUNK; ++i)       // i = 36..39: last col may be halo
        emit(i, (i == WCHUNK - 1) ? colG(ws + i + 1) : colF(ws + i + 1));
}

extern "C" void kernel_launch(void* const* d_in, const int* in_sizes, int n_in,
                              void* d_out, int out_size, void* d_ws, size_t ws_size,
                              hipStream_t stream) {
    (void)in_sizes; (void)n_in; (void)out_size; (void)d_ws; (void)ws_size;
    const float* x = (const float*)d_in[0];
    float* out = (float*)d_out;
    const int nblocks = BC * NTILES_D * NTILES_H;  // 1600
    sobel3d_mag_kernel<<<dim3(nblocks), dim3(256), 0, stream>>>(x, out);
}